// Net_65712999629401
// MI455X (gfx1250) — compile-verified
//
#include <hip/hip_runtime.h>
#include <hip/hip_bf16.h>

typedef __attribute__((ext_vector_type(2))) float v2f;
typedef __attribute__((ext_vector_type(8))) float v8f;

#define BN_EPS 1e-5f

// ---------------------------------------------------------------------------
// AdderNet "conv": out[b,o,ho,wo] = -sum_{c,kh,kw} |x[b,c,ho+kh,wo+kw] - w[o,c,kh,kw]|
// blockIdx.y = output channel o; its weight slice (C*K*K <= 500 floats) cached in LDS.
// ---------------------------------------------------------------------------
__global__ void adder_conv_kernel(const float* __restrict__ x,
                                  const float* __restrict__ w,
                                  float* __restrict__ out,
                                  int B, int C, int H, int W,
                                  int O, int K, int Ho, int Wo) {
    __shared__ float wsm[512];
    const int o = blockIdx.y;
    const int ckk = C * K * K;
    for (int i = threadIdx.x; i < ckk; i += blockDim.x)
        wsm[i] = w[(size_t)o * ckk + i];
    __syncthreads();

    const int total = B * Ho * Wo;
    for (int idx = blockIdx.x * blockDim.x + threadIdx.x; idx < total;
         idx += gridDim.x * blockDim.x) {
        int wo = idx % Wo;
        int t  = idx / Wo;
        int ho = t % Ho;
        int b  = t / Ho;
        const float* xb = x + (size_t)b * C * H * W;
        float acc = 0.0f;
        int wi = 0;
        for (int c = 0; c < C; ++c) {
            const float* xc = xb + (size_t)c * H * W + ho * W + wo;
            #pragma unroll
            for (int kh = 0; kh < 5; ++kh) {
                #pragma unroll
                for (int kw = 0; kw < 5; ++kw) {
                    acc += fabsf(xc[kh * W + kw] - wsm[wi++]);
                }
            }
        }
        out[((size_t)b * O + o) * (Ho * Wo) + ho * Wo + wo] = -acc;
    }
}

// ---------------------------------------------------------------------------
// Per-channel batch statistics (mean, 1/sqrt(biased var + eps)); one block/channel.
// ---------------------------------------------------------------------------
__global__ void bn_stats_kernel(const float* __restrict__ h,
                                float* __restrict__ mean,
                                float* __restrict__ invstd,
                                int B, int C, int HW) {
    const int c = blockIdx.x;
    const int total = B * HW;
    float s = 0.0f, ss = 0.0f;
    for (int i = threadIdx.x; i < total; i += blockDim.x) {
        int b  = i / HW;
        int hw = i - b * HW;
        float v = h[((size_t)b * C + c) * HW + hw];
        s  += v;
        ss += v * v;
    }
    __shared__ float rs[256];
    __shared__ float rss[256];
    rs[threadIdx.x]  = s;
    rss[threadIdx.x] = ss;
    __syncthreads();
    for (int off = blockDim.x >> 1; off > 0; off >>= 1) {
        if ((int)threadIdx.x < off) {
            rs[threadIdx.x]  += rs[threadIdx.x + off];
            rss[threadIdx.x] += rss[threadIdx.x + off];
        }
        __syncthreads();
    }
    if (threadIdx.x == 0) {
        float m   = rs[0] / (float)total;
        float var = rss[0] / (float)total - m * m;
        mean[c]   = m;
        invstd[c] = rsqrtf(var + BN_EPS);
    }
}

// ---------------------------------------------------------------------------
// Fused BatchNorm (affine) + 2x2 maxpool (stride 2).
// ---------------------------------------------------------------------------
__global__ void bn_maxpool_kernel(const float* __restrict__ h,
                                  const float* __restrict__ mean,
                                  const float* __restrict__ invstd,
                                  const float* __restrict__ gamma,
                                  const float* __restrict__ beta,
                                  float* __restrict__ out,
                                  int B, int C, int H, int W, int total) {
    const int Ho = H >> 1, Wo = W >> 1;
    int idx = blockIdx.x * blockDim.x + threadIdx.x;
    if (idx >= total) return;
    int wo = idx % Wo;
    int t  = idx / Wo;
    int ho = t % Ho;
    t /= Ho;
    int c = t % C;
    int b = t / C;
    const float* p = h + ((size_t)b * C + c) * H * W + (ho * 2) * W + wo * 2;
    float g  = gamma[c] * invstd[c];
    float bb = beta[c] - mean[c] * g;
    float v0 = fmaf(p[0],     g, bb);
    float v1 = fmaf(p[1],     g, bb);
    float v2 = fmaf(p[W],     g, bb);
    float v3 = fmaf(p[W + 1], g, bb);
    out[idx] = fmaxf(fmaxf(v0, v1), fmaxf(v2, v3));
}

// ---------------------------------------------------------------------------
// Dense layer via V_WMMA_F32_16X16X4_F32.
//   out[M,N] = A[M,K] * Wt[N,K]^T + bias[N]   (optional ReLU)
// One wave (32 lanes) per block computes one 16x16 output tile.
// A fragment layout (ISA 7.12.2, 32-bit A 16x4): lane&15 = row M,
// lane>>4 selects K pair {0,1} vs {2,3}. B mirrors with N = lane&15.
// C/D: vgpr v -> row = v + 8*(lane>>4), col = lane&15.
//
// N-edge handling: output column n depends only on column n of B, so lanes
// whose column is out of range just read a clamped (valid) row of Wt and
// accumulate garbage that is never stored. No masking in the inner loop:
// steady state is pure global_load_b64 + v_wmma.
// M and K must be multiples of 16 / 4 (true here: M=128, K in {800,500}).
// ---------------------------------------------------------------------------
__global__ void wmma_fc_kernel(const float* __restrict__ A,
                               const float* __restrict__ Wt,
                               const float* __restrict__ bias,
                               float* __restrict__ out,
                               int M, int N, int K, int doRelu) {
    const int mBase = blockIdx.x * 16;
    const int nBase = blockIdx.y * 16;
    const int lane  = threadIdx.x;      // 0..31, full wave32
    const int half  = lane >> 4;        // 0 or 1
    const int l     = lane & 15;

    const int aRow  = mBase + l;
    const int nCol  = nBase + l;
    const bool colOk = (nCol < N);

    const float* aPtr = A  + (size_t)aRow * K + half * 2;
    const float* bPtr = Wt + (size_t)(colOk ? nCol : 0) * K + half * 2;

    v8f c = {};
    for (int k = 0; k < K; k += 4) {
        v2f a, b;
        a.x = aPtr[k];
        a.y = aPtr[k + 1];
        b.x = bPtr[k];
        b.y = bPtr[k + 1];
        c = __builtin_amdgcn_wmma_f32_16x16x4_f32(
                /*neg_a=*/false, a, /*neg_b=*/false, b,
                /*c_mod=*/(short)0, c, /*reuse_a=*/false, /*reuse_b=*/false);
    }

    if (colOk) {
        float bv = bias[nCol];
        #pragma unroll
        for (int v = 0; v < 8; ++v) {
            int row = mBase + v + half * 8;
            float val = c[v] + bv;
            if (doRelu) val = fmaxf(val, 0.0f);
            out[(size_t)row * N + nCol] = val;
        }
    }
}

// ---------------------------------------------------------------------------
// Row-wise softmax in place, one thread per row (N=10).
// ---------------------------------------------------------------------------
__global__ void softmax_kernel(float* __restrict__ x, int M, int N) {
    int row = blockIdx.x * blockDim.x + threadIdx.x;
    if (row >= M) return;
    float* p = x + (size_t)row * N;
    float mx = p[0];
    for (int i = 1; i < N; ++i) mx = fmaxf(mx, p[i]);
    float s = 0.0f;
    for (int i = 0; i < N; ++i) {
        float e = expf(p[i] - mx);
        p[i] = e;
        s += e;
    }
    float inv = 1.0f / s;
    for (int i = 0; i < N; ++i) p[i] *= inv;
}

// ---------------------------------------------------------------------------
extern "C" void kernel_launch(void* const* d_in, const int* in_sizes, int n_in,
                              void* d_out, int out_size, void* d_ws, size_t ws_size,
                              hipStream_t stream) {
    const float* x      = (const float*)d_in[0];   // [128,1,28,28]
    const float* w1     = (const float*)d_in[1];   // [20,1,5,5]
    const float* gamma1 = (const float*)d_in[2];   // [20]
    const float* beta1  = (const float*)d_in[3];   // [20]
    const float* w2     = (const float*)d_in[4];   // [50,20,5,5]
    const float* gamma2 = (const float*)d_in[5];   // [50]
    const float* beta2  = (const float*)d_in[6];   // [50]
    const float* fc1_w  = (const float*)d_in[7];   // [500,800]
    const float* fc1_b  = (const float*)d_in[8];   // [500]
    const float* fc2_w  = (const float*)d_in[9];   // [10,500]
    const float* fc2_b  = (const float*)d_in[10];  // [10]
    float* out = (float*)d_out;                    // [128,10]

    const int B = 128;

    // Workspace layout (floats), total ~9.7 MB
    float* ws = (float*)d_ws;
    float* h1 = ws;                    // [128,20,24,24] = 1,474,560
    float* p1 = h1 + 1474560;          // [128,20,12,12] =   368,640
    float* h2 = p1 + 368640;           // [128,50, 8, 8] =   409,600
    float* p2 = h2 + 409600;           // [128,50, 4, 4] =   102,400  -> [128,800]
    float* h3 = p2 + 102400;           // [128,500]      =    64,000
    float* mean1 = h3 + 64000;         // [20]
    float* inv1  = mean1 + 20;         // [20]
    float* mean2 = inv1 + 20;          // [50]
    float* inv2  = mean2 + 50;         // [50]

    // ---- Stage 1: adder conv1 [128,1,28,28] -> [128,20,24,24] ----
    {
        dim3 grid(64, 20);
        adder_conv_kernel<<<grid, 256, 0, stream>>>(x, w1, h1,
                                                    B, 1, 28, 28, 20, 5, 24, 24);
    }
    bn_stats_kernel<<<20, 256, 0, stream>>>(h1, mean1, inv1, B, 20, 24 * 24);
    {
        int total = B * 20 * 12 * 12;
        bn_maxpool_kernel<<<(total + 255) / 256, 256, 0, stream>>>(
            h1, mean1, inv1, gamma1, beta1, p1, B, 20, 24, 24, total);
    }

    // ---- Stage 2: adder conv2 [128,20,12,12] -> [128,50,8,8] ----
    {
        dim3 grid(32, 50);
        adder_conv_kernel<<<grid, 256, 0, stream>>>(p1, w2, h2,
                                                    B, 20, 12, 12, 50, 5, 8, 8);
    }
    bn_stats_kernel<<<50, 256, 0, stream>>>(h2, mean2, inv2, B, 50, 8 * 8);
    {
        int total = B * 50 * 4 * 4;
        bn_maxpool_kernel<<<(total + 255) / 256, 256, 0, stream>>>(
            h2, mean2, inv2, gamma2, beta2, p2, B, 50, 8, 8, total);
    }

    // ---- FC1: [128,800] x [800,500]^T + b, ReLU  (WMMA f32 16x16x4) ----
    {
        dim3 grid(128 / 16, (500 + 15) / 16);   // (8, 32)
        wmma_fc_kernel<<<grid, 32, 0, stream>>>(p2, fc1_w, fc1_b, h3,
                                                128, 500, 800, /*relu=*/1);
    }

    // ---- FC2: [128,500] x [500,10]^T + b -> logits in d_out ----
    {
        dim3 grid(128 / 16, (10 + 15) / 16);    // (8, 1)
        wmma_fc_kernel<<<grid, 32, 0, stream>>>(h3, fc2_w, fc2_b, out,
                                                128, 10, 500, /*relu=*/0);
    }

    // ---- Softmax in place on d_out ----
    softmax_kernel<<<1, 128, 0, stream>>>(out, 128, 10);
}